// DGPMotionEstimator_31610959298965
// MI455X (gfx1250) — compile-verified
//
#include <hip/hip_runtime.h>
#include <math.h>

#define Tt 50
#define Pp 1000
#define Ss 10
#define Mm 100
#define Nn 4000
#define Dd 2
#define PS (Pp * Ss)          // 10000
#define MPAD 112              // 7 * 16, zero-padded M
#define NKT (MPAD / 4)        // 28 K-steps of 4
#define NRT (MPAD / 16)       // 7 row tiles
#define CPW 16                // columns per wave
#define WPB 4                 // waves per block
#define CPB (CPW * WPB)       // 64 columns per block
#define JITTERF 1e-5f
#define WSTRIDE (MPAD * MPAD) // floats per per-step W matrix

typedef float v2f __attribute__((ext_vector_type(2)));
typedef float v8f __attribute__((ext_vector_type(8)));

// ---------------------------------------------------------------------------
// samples[0] = mu0 + exp(log_sigma0) * eps0
// ---------------------------------------------------------------------------
__global__ void init_kernel(const float* __restrict__ mu0,
                            const float* __restrict__ ls0,
                            const float* __restrict__ eps0,
                            float* __restrict__ samples) {
  int idx = blockIdx.x * blockDim.x + threadIdx.x;
  if (idx >= PS * Dd) return;
  int d  = idx & 1;
  int ps = idx >> 1;        // p*S + s
  int p  = ps / Ss;
  samples[idx] = mu0[p * Dd + d] + __expf(ls0[p * Dd + d]) * eps0[idx];
}

// ---------------------------------------------------------------------------
// One workgroup per time step t = blockIdx.x (all 49 run concurrently):
// Kzz -> Cholesky -> Linv -> W_t = Linv^T Linv (zero-padded 112x112).
// Kzz depends only on kernel inputs, never on the sampled trajectory,
// so the whole batch is independent of the sequential recurrence.
// ---------------------------------------------------------------------------
__global__ __launch_bounds__(128)
void chol_inv_kernel(const float* __restrict__ Zall,   // [T-1, M, D]
                     const float* __restrict__ lls,
                     const float* __restrict__ lvar,
                     float* __restrict__ Wall) {       // [T-1, 112*112]
  __shared__ float A[Mm][Mm];
  __shared__ float Li[Mm][Mm];
  __shared__ float z[Mm][Dd];
  const int t   = blockIdx.x;
  const int tid = threadIdx.x;
  const float* Z = Zall + (size_t)t * Mm * Dd;
  float* W       = Wall + (size_t)t * WSTRIDE;
  const float ls2 = __expf(2.f * lls[t]);
  const float kv  = __expf(lvar[t]);

  for (int i = tid; i < Mm * Dd; i += 128) ((float*)z)[i] = Z[i];
  __syncthreads();

  for (int e = tid; e < Mm * Mm; e += 128) {
    int i = e / Mm, j = e % Mm;
    float dx = z[i][0] - z[j][0];
    float dy = z[i][1] - z[j][1];
    float v = kv * __expf(-0.5f * (dx * dx + dy * dy) / ls2);
    if (i == j) v += JITTERF;
    A[i][j] = v;
  }
  __syncthreads();

  // right-looking Cholesky (lower) in LDS
  for (int k = 0; k < Mm; ++k) {
    if (tid == 0) A[k][k] = sqrtf(A[k][k]);
    __syncthreads();
    for (int i = k + 1 + tid; i < Mm; i += 128) A[i][k] /= A[k][k];
    __syncthreads();
    for (int i = k + 1 + tid; i < Mm; i += 128) {
      float aik = A[i][k];
      for (int j = k + 1; j <= i; ++j) A[i][j] -= aik * A[j][k];
    }
    __syncthreads();
  }

  // Linv: forward substitution, one column per lane
  if (tid < Mm) {
    int j = tid;
    for (int i = 0; i < j; ++i) Li[i][j] = 0.f;
    Li[j][j] = 1.f / A[j][j];
    for (int i = j + 1; i < Mm; ++i) {
      float s = 0.f;
      for (int k2 = j; k2 < i; ++k2) s += A[i][k2] * Li[k2][j];
      Li[i][j] = -s / A[i][i];
    }
  }
  __syncthreads();

  // W = Linv^T Linv, zero-padded to 112x112
  for (int e = tid; e < MPAD * MPAD; e += 128) {
    int i = e / MPAD, j = e % MPAD;
    float s = 0.f;
    if (i < Mm && j < Mm) {
      int k0 = i > j ? i : j;
      for (int k2 = k0; k2 < Mm; ++k2) s += Li[k2][i] * Li[k2][j];
    }
    W[e] = s;
  }
}

// ---------------------------------------------------------------------------
// Heavy kernel: per 64 columns, build k slab in LDS, B = W @ k via
// v_wmma_f32_16x16x4_f32, then mean/qvar/var reductions and residual update.
// ---------------------------------------------------------------------------
__global__ __launch_bounds__(128)
void step_kernel(const float* __restrict__ Wall,     // [T-1, 112*112]
                 float* __restrict__ samples,        // [T, PS, D]
                 int t,
                 const float* __restrict__ Z,        // [M, D]  (offset by t)
                 const float* __restrict__ um_g,     // [M, D]
                 const float* __restrict__ uls_g,    // [M, D]
                 const float* __restrict__ lls,
                 const float* __restrict__ lvar,
                 const float* __restrict__ eps) {    // [PS, D] (offset by t)
  __shared__ float kmat[MPAD * CPB];   // 112 x 64 column slab of Kzx
  __shared__ float xs[CPB][Dd];
  __shared__ float z[Mm][Dd];
  __shared__ float um[MPAD][Dd];
  __shared__ float su[MPAD][Dd];

  const int tid  = threadIdx.x;
  const int base = blockIdx.x * CPB;
  const float ls2 = __expf(2.f * lls[t]);
  const float kv  = __expf(lvar[t]);
  const float* W    = Wall + (size_t)t * WSTRIDE;
  const float* prev = samples + (size_t)t * PS * Dd;
  float* nxt        = samples + (size_t)(t + 1) * PS * Dd;

  for (int i = tid; i < Mm * Dd; i += 128) ((float*)z)[i] = Z[i];
  for (int i = tid; i < MPAD * Dd; i += 128) {
    int m = i >> 1;
    ((float*)um)[i] = (m < Mm) ? um_g[i] : 0.f;
    ((float*)su)[i] = (m < Mm) ? __expf(2.f * uls_g[i]) : 0.f;
  }
  for (int i = tid; i < CPB * Dd; i += 128) {
    int c = i >> 1, d = i & 1;
    int cg = base + c;
    xs[c][d] = (cg < PS) ? prev[cg * Dd + d] : 0.f;
  }
  __syncthreads();

  for (int e = tid; e < MPAD * CPB; e += 128) {
    int m = e >> 6, c = e & 63;
    float v = 0.f;
    if (m < Mm) {
      float dx = z[m][0] - xs[c][0];
      float dy = z[m][1] - xs[c][1];
      v = kv * __expf(-0.5f * (dx * dx + dy * dy) / ls2);
    }
    kmat[e] = v;
  }
  __syncthreads();

  const int lane   = tid & 31;
  const int wv     = tid >> 5;
  const int laneLo = lane & 15;
  const int laneHi = lane >> 4;
  const int cloc   = wv * CPW + laneLo;

  v8f acc[NRT];
#pragma unroll
  for (int r = 0; r < NRT; ++r) {
#pragma unroll
    for (int j = 0; j < 8; ++j) acc[r][j] = 0.f;
  }

  // D[112x16] = W[112x112] x k[112x16]; A frag K = kk*4 + vgpr + 2*laneHi
  for (int kk = 0; kk < NKT; ++kk) {
    const int kbase = kk * 4 + 2 * laneHi;
    v2f bfrag;
    bfrag.x = kmat[(kbase + 0) * CPB + cloc];
    bfrag.y = kmat[(kbase + 1) * CPB + cloc];
#pragma unroll
    for (int r = 0; r < NRT; ++r) {
      const int row = r * 16 + laneLo;
      v2f afrag = *(const v2f*)(W + row * MPAD + kbase);
      acc[r] = __builtin_amdgcn_wmma_f32_16x16x4_f32(
          false, afrag, false, bfrag, (short)0, acc[r], false, false);
    }
  }

  // per-column reductions straight out of the C-layout:
  // lane holds rows r*16 + j + 8*laneHi of column cloc
  float m0 = 0.f, m1 = 0.f, qk = 0.f, v0 = 0.f, v1 = 0.f;
#pragma unroll
  for (int r = 0; r < NRT; ++r) {
#pragma unroll
    for (int j = 0; j < 8; ++j) {
      int row = r * 16 + j + 8 * laneHi;
      float b = acc[r][j];
      m0 += b * um[row][0];
      m1 += b * um[row][1];
      qk += b * kmat[row * CPB + cloc];
      v0 += b * b * su[row][0];
      v1 += b * b * su[row][1];
    }
  }
  m0 += __shfl_xor(m0, 16, 32);
  m1 += __shfl_xor(m1, 16, 32);
  qk += __shfl_xor(qk, 16, 32);
  v0 += __shfl_xor(v0, 16, 32);
  v1 += __shfl_xor(v1, 16, 32);

  int cg = base + cloc;
  if (laneHi == 0 && cg < PS) {
    float qvar = kv - qk;
    float var0 = fmaxf(qvar + v0, 1e-8f);
    float var1 = fmaxf(qvar + v1, 1e-8f);
    float e0 = eps[cg * Dd + 0];
    float e1 = eps[cg * Dd + 1];
    nxt[cg * Dd + 0] = xs[cloc][0] + m0 + sqrtf(var0) * e0;
    nxt[cg * Dd + 1] = xs[cloc][1] + m1 + sqrtf(var1) * e1;
  }
}

// ---------------------------------------------------------------------------
// out[n, p] = mean_s logN(data[n] | samples[time[n], p, s], sp^2 I)
// ---------------------------------------------------------------------------
__global__ void logp_kernel(const float* __restrict__ data,
                            const int* __restrict__ time,
                            const float* __restrict__ samples,
                            const float* __restrict__ noise_p,
                            float* __restrict__ out) {
  int idx = blockIdx.x * blockDim.x + threadIdx.x;
  if (idx >= Nn * Pp) return;
  int n = idx / Pp;
  int p = idx % Pp;
  float sp  = log1pf(__expf(noise_p[0]));   // softplus
  float c   = -0.5f * Dd * logf(2.f * 3.14159265358979f) - Dd * logf(sp);
  float inv = 0.5f / (sp * sp);
  int t = time[n];
  const float* b = samples + ((size_t)t * PS + (size_t)p * Ss) * Dd;
  float dx0 = data[n * Dd + 0];
  float dx1 = data[n * Dd + 1];
  float s = 0.f;
#pragma unroll
  for (int si = 0; si < Ss; ++si) {
    float d0 = dx0 - b[si * 2 + 0];
    float d1 = dx1 - b[si * 2 + 1];
    s += c - inv * (d0 * d0 + d1 * d1);
  }
  out[idx] = s * (1.f / Ss);
}

// ---------------------------------------------------------------------------
extern "C" void kernel_launch(void* const* d_in, const int* in_sizes, int n_in,
                              void* d_out, int out_size, void* d_ws, size_t ws_size,
                              hipStream_t stream) {
  (void)in_sizes; (void)n_in; (void)out_size; (void)ws_size;
  const float* data  = (const float*)d_in[0];
  const int*   time  = (const int*)d_in[1];
  const float* mu0   = (const float*)d_in[2];
  const float* ls0   = (const float*)d_in[3];
  const float* Z     = (const float*)d_in[4];
  const float* umean = (const float*)d_in[5];
  const float* ulog  = (const float*)d_in[6];
  const float* lls   = (const float*)d_in[7];
  const float* lvar  = (const float*)d_in[8];
  const float* noise = (const float*)d_in[9];
  const float* eps0  = (const float*)d_in[10];
  const float* epst  = (const float*)d_in[11];

  float* Wall    = (float*)d_ws;              // (T-1)*112*112 = 614,656 floats
  float* samples = (float*)d_ws + 1048576;    // T*PS*D = 1,000,000 floats

  // All 49 Cholesky factorizations + inversions are independent of the
  // trajectory recurrence -> batch them in one concurrent launch.
  chol_inv_kernel<<<Tt - 1, 128, 0, stream>>>(Z, lls, lvar, Wall);

  init_kernel<<<(PS * Dd + 255) / 256, 256, 0, stream>>>(mu0, ls0, eps0, samples);

  for (int t = 0; t < Tt - 1; ++t) {
    step_kernel<<<(PS + CPB - 1) / CPB, 128, 0, stream>>>(
        Wall, samples, t,
        Z + (size_t)t * Mm * Dd,
        umean + (size_t)t * Mm * Dd,
        ulog + (size_t)t * Mm * Dd,
        lls, lvar,
        epst + (size_t)t * PS * Dd);
  }

  logp_kernel<<<(Nn * Pp + 255) / 256, 256, 0, stream>>>(
      data, time, samples, noise, (float*)d_out);
}